// NeighborList_8916352106876
// MI455X (gfx1250) — compile-verified
//
#include <hip/hip_runtime.h>

#define N_ATOMS   4096
#define MAXP      131072
#define NTILES    (N_ATOMS / 16)      // 256 column tiles of 16
#define CUTF      5.0f

typedef float v2f __attribute__((ext_vector_type(2)));
typedef float v8f __attribute__((ext_vector_type(8)));
typedef int   v4i __attribute__((ext_vector_type(4)));

typedef __attribute__((address_space(1))) v4i gv4i;   // global int4
typedef __attribute__((address_space(3))) v4i lv4i;   // LDS int4

// Exact min-image squared distance (round-half-even like jnp.round).
// Used identically in count (borderline/slow path) and fill so both agree.
__device__ __forceinline__ float wrapped_d2(float xi, float yi, float zi,
                                            float xj, float yj, float zj,
                                            float Lx, float Ly, float Lz,
                                            float iLx, float iLy, float iLz,
                                            float& dx, float& dy, float& dz) {
    dx = xi - xj; dy = yi - yj; dz = zi - zj;
    dx -= Lx * rintf(dx * iLx);
    dy -= Ly * rintf(dy * iLy);
    dz -= Lz * rintf(dz * iLz);
    return dx * dx + dy * dy + dz * dz;
}

// Async memory->LDS staging (CDNA5 path), with fallbacks.
__device__ __forceinline__ void stage_pos_to_lds(const float* __restrict__ pos,
                                                 float* sP, int tid) {
#if __has_builtin(__builtin_amdgcn_global_load_async_to_lds_b128)
    for (int t = tid; t < (3 * N_ATOMS) / 4; t += 256) {
        __builtin_amdgcn_global_load_async_to_lds_b128(
            (gv4i*)pos + t, (lv4i*)sP + t, 0, 0);
    }
  #if __has_builtin(__builtin_amdgcn_s_wait_asynccnt)
    __builtin_amdgcn_s_wait_asynccnt(0);
  #else
    asm volatile("s_wait_asynccnt 0x0" ::: "memory");
  #endif
#else
    // fallback: plain vectorized copy through VGPRs
    for (int t = tid; t < (3 * N_ATOMS) / 4; t += 256)
        ((float4*)sP)[t] = ((const float4*)pos)[t];
#endif
}

// ---------------------------------------------------------------------------
// Pass 0: initialize output padding (pairs = -1, deltas/dists = 0)
// ---------------------------------------------------------------------------
__global__ __launch_bounds__(256) void nl_init(int* outInt, float* outF) {
    int idx = blockIdx.x * 256 + threadIdx.x;
    if (idx < 2 * MAXP)           outInt[idx] = -1;       // pair_i, pair_j pad
    else if (idx < 6 * MAXP)      outF[idx]   = 0.0f;     // deltas + distances pad
}

// ---------------------------------------------------------------------------
// Pass 1: per-row hit counts via WMMA raw-distance classifier.
// One workgroup per 16-row strip; 8 waves split the 256 column tiles.
// LDS holds the raw AoS position image (48 KB), staged asynchronously.
// ---------------------------------------------------------------------------
__global__ __launch_bounds__(256) void nl_count(const float* __restrict__ pos,
                                                const float* __restrict__ box,
                                                int* __restrict__ rowCnt) {
    __shared__ float sP[3 * N_ATOMS];   // AoS copy of positions, 48 KB
    __shared__ int   sCnt[16];

    const int tid  = threadIdx.x;
    const int lane = tid & 31;
    const int wave = tid >> 5;
    const int half = lane >> 4;      // 0: lanes 0-15, 1: lanes 16-31
    const int ln   = lane & 15;

    stage_pos_to_lds(pos, sP, tid);
    if (tid < 16) sCnt[tid] = 0;
    __syncthreads();

    const float Lx = box[0], Ly = box[4], Lz = box[8];
    const float iLx = 1.0f / Lx, iLy = 1.0f / Ly, iLz = 1.0f / Lz;
    const float t1   = CUTF * CUTF;                    // 25
    const float Lmin = fminf(Lx, fminf(Ly, Lz));
    const float t2   = (Lmin - CUTF) * (Lmin - CUTF);  // 1225: below this no wrap
    const float eps  = 1.0e-2f;                        // WMMA rounding guard band
    const float t1me = t1 - eps, t1pe = t1 + eps;

    const int strip = blockIdx.x;      // 0..255
    const int ib    = strip * 16;

    // A-matrix (16x4 f32): row i = ib+M, features K0=-2x K1=-2y K2=-2z K3=|r|^2
    // Layout rule: row K = vgpr + 2*half (ISA 16x4 table).
    const float xi0 = sP[3 * (ib + ln) + 0];
    const float yi0 = sP[3 * (ib + ln) + 1];
    const float zi0 = sP[3 * (ib + ln) + 2];
    const float qi0 = xi0 * xi0 + yi0 * yi0 + zi0 * zi0;
    v2f amat;
    amat[0] = half ? (-2.0f * zi0) : (-2.0f * xi0);   // K2 / K0
    amat[1] = half ? qi0           : (-2.0f * yi0);   // K3 / K1

    // Loop-invariant: this lane's 8 candidate i-rows (row = ib + v + 8*half).
    float xr[8], yr[8], zr[8];
#pragma unroll
    for (int v = 0; v < 8; ++v) {
        const int r = ib + v + (half << 3);
        xr[v] = sP[3 * r + 0];
        yr[v] = sP[3 * r + 1];
        zr[v] = sP[3 * r + 2];
    }

    int cnt[8];
#pragma unroll
    for (int v = 0; v < 8; ++v) cnt[v] = 0;

    for (int jt = strip + wave; jt < NTILES; jt += 8) {
        const int jb = jt * 16;
        const int j  = jb + ln;
        const float xj = sP[3 * j + 0];
        const float yj = sP[3 * j + 1];
        const float zj = sP[3 * j + 2];
        const float qj = xj * xj + yj * yj + zj * zj;

        // B-matrix (4x16 f32): col j = jb+N, rows K0=x K1=y K2=z K3=1
        v2f bmat;
        bmat[0] = half ? zj   : xj;    // K2 / K0
        bmat[1] = half ? 1.0f : yj;    // K3 / K1

        v8f cmat;
#pragma unroll
        for (int v = 0; v < 8; ++v) cmat[v] = qj;     // + |r_j|^2 per column

        // raw_d2[m][n] = |r_i|^2 + |r_j|^2 - 2 r_i . r_j
        v8f dmat = __builtin_amdgcn_wmma_f32_16x16x4_f32(
            false, amat, false, bmat, (short)0, cmat, false, false);

#pragma unroll
        for (int v = 0; v < 8; ++v) {
            const int i = ib + v + (half << 3);
            const float raw = dmat[v];
            const bool valid = (j > i);
            bool hit = valid && (raw < t1me);          // definite hit, no wrap
            const bool needSlow =
                valid && !(raw < t1me) && ((raw < t1pe) || (raw >= t2));
            // Wave-uniform skip: only run the exact wrap path if some lane needs it.
            if (__builtin_amdgcn_ballot_w32(needSlow) != 0u) {
                float dx, dy, dz;
                const float d2 = wrapped_d2(xr[v], yr[v], zr[v], xj, yj, zj,
                                            Lx, Ly, Lz, iLx, iLy, iLz, dx, dy, dz);
                hit = hit || (needSlow && (d2 < t1));
            }
            cnt[v] += hit ? 1 : 0;
        }
    }

    // One-time LDS accumulation: this lane's vgpr v maps to row v + 8*half.
#pragma unroll
    for (int v = 0; v < 8; ++v)
        if (cnt[v]) atomicAdd(&sCnt[v + (half << 3)], cnt[v]);

    __syncthreads();
    if (tid < 16) rowCnt[ib + tid] = sCnt[tid];
}

// ---------------------------------------------------------------------------
// Pass 2: exclusive scan of 4096 row counts (single workgroup); total pairs.
// ---------------------------------------------------------------------------
__global__ __launch_bounds__(256) void nl_scan(const int* __restrict__ rowCnt,
                                               int* __restrict__ rowOff,
                                               int* __restrict__ nPairsOut) {
    __shared__ int sSum[256];
    const int tid  = threadIdx.x;
    const int base = tid * 16;

    int local[16];
    int s = 0;
#pragma unroll
    for (int k = 0; k < 16; ++k) { local[k] = s; s += rowCnt[base + k]; }
    sSum[tid] = s;
    __syncthreads();

    for (int off = 1; off < 256; off <<= 1) {
        const int addend = (tid >= off) ? sSum[tid - off] : 0;
        __syncthreads();
        sSum[tid] += addend;
        __syncthreads();
    }

    const int prev = (tid == 0) ? 0 : sSum[tid - 1];
#pragma unroll
    for (int k = 0; k < 16; ++k) rowOff[base + k] = prev + local[k];
    if (tid == 255) nPairsOut[0] = sSum[255];
}

// ---------------------------------------------------------------------------
// Pass 3: one wave per row i; write hits in ascending-j order at rowOff[i].
// ---------------------------------------------------------------------------
__global__ __launch_bounds__(256) void nl_fill(const float* __restrict__ pos,
                                               const float* __restrict__ box,
                                               const int* __restrict__ rowOff,
                                               int* __restrict__ pairI,
                                               int* __restrict__ pairJ,
                                               float* __restrict__ deltas,
                                               float* __restrict__ dists) {
    const int lane = threadIdx.x & 31;
    const int i    = blockIdx.x * 8 + (threadIdx.x >> 5);   // 512 blocks * 8 waves

    const float Lx = box[0], Ly = box[4], Lz = box[8];
    const float iLx = 1.0f / Lx, iLy = 1.0f / Ly, iLz = 1.0f / Lz;
    const float t1 = CUTF * CUTF;

    const float xi = pos[3 * i + 0];
    const float yi = pos[3 * i + 1];
    const float zi = pos[3 * i + 2];

    const int base = rowOff[i];
    int written = 0;

    for (int j0 = i + 1; j0 < N_ATOMS; j0 += 32) {
        const int j = j0 + lane;
        bool ok = false;
        float dx = 0.f, dy = 0.f, dz = 0.f, d2 = 0.f;
        if (j < N_ATOMS) {
            d2 = wrapped_d2(xi, yi, zi,
                            pos[3 * j], pos[3 * j + 1], pos[3 * j + 2],
                            Lx, Ly, Lz, iLx, iLy, iLz, dx, dy, dz);
            ok = d2 < t1;
        }
        const unsigned msk = __builtin_amdgcn_ballot_w32(ok);
        if (ok) {
            const int slot = base + written + __popc(msk & ((1u << lane) - 1u));
            if (slot < MAXP) {
                pairI[slot] = i;
                pairJ[slot] = j;
                deltas[3 * slot + 0] = dx;
                deltas[3 * slot + 1] = dy;
                deltas[3 * slot + 2] = dz;
                dists[slot] = sqrtf(d2);
            }
        }
        written += __popc(msk);
    }
}

// ---------------------------------------------------------------------------
extern "C" void kernel_launch(void* const* d_in, const int* in_sizes, int n_in,
                              void* d_out, int out_size, void* d_ws, size_t ws_size,
                              hipStream_t stream) {
    const float* pos = (const float*)d_in[0];   // [4096, 3] f32
    const float* box = (const float*)d_in[1];   // [3, 3] f32 (diagonal used)

    int* rowCnt = (int*)d_ws;                   // 4096 ints
    int* rowOff = rowCnt + N_ATOMS;             // 4096 ints

    int*   outInt = (int*)d_out;
    float* outF   = (float*)d_out;
    // layout (element offsets): pairs_i [0,MAXP), pairs_j [MAXP,2*MAXP),
    // deltas [2*MAXP, 5*MAXP), distances [5*MAXP, 6*MAXP), n_pairs at 6*MAXP.

    nl_init <<<(6 * MAXP + 255) / 256, 256, 0, stream>>>(outInt, outF);
    nl_count<<<NTILES, 256, 0, stream>>>(pos, box, rowCnt);
    nl_scan <<<1, 256, 0, stream>>>(rowCnt, rowOff, outInt + 6 * MAXP);
    nl_fill <<<N_ATOMS / 8, 256, 0, stream>>>(pos, box, rowOff,
                                              outInt, outInt + MAXP,
                                              outF + 2 * MAXP, outF + 5 * MAXP);
}